// RunLengthEventSetEmbedding_50053548867748
// MI455X (gfx1250) — compile-verified
//
#include <hip/hip_runtime.h>

#define NTIME 2048
#define NCOMP 8
#define HID 64
#define OUTD 64
#define BATCH 64
#define MAX_EVENTS_F 16384.0f
#define AP 66   // LDS row pitch (floats) to dodge bank conflicts on column reads

typedef __attribute__((ext_vector_type(2))) float v2f;
typedef __attribute__((ext_vector_type(8))) float v8f;

// ---------------------------------------------------------------------------
// Kernel 1: run-length event extraction fused with layer-1 MLP + pooling sum.
// One block per batch element; one wave32 per component. Produces
//   S[b][u]   = sum_events relu(W1[c,u] + s*W1[8,u] + sn*W1[9,u] + d*W1[10,u] + b1[u])
//   valid[b]  = max(#events, 1)
// ---------------------------------------------------------------------------
__global__ __launch_bounds__(256) void event_scan_kernel(
    const float* __restrict__ x, const float* __restrict__ W1,
    const float* __restrict__ b1, float* __restrict__ S,
    float* __restrict__ validv)
{
    __shared__ unsigned char seqb[NCOMP * NTIME];   // 16 KB bit sequence (bytes)
    __shared__ float accLDS[NCOMP][HID];
    __shared__ int   cntLDS[NCOMP];

    const int b   = blockIdx.x;
    const int tid = threadIdx.x;
    const float* xb = x + (size_t)b * NTIME * NCOMP;

    // Coalesced load of x[b]; threshold to bits, transpose to [c][t] bytes.
    for (int f = tid; f < NTIME * NCOMP; f += 256) {
        int t = f >> 3, c = f & 7;
        seqb[c * NTIME + t] = (xb[f] > 0.5f) ? 1 : 0;
    }
    __syncthreads();

    const int c    = tid >> 5;   // wave id == component
    const int lane = tid & 31;
    const int u0 = lane, u1 = lane + 32;   // each lane owns 2 hidden units

    // Preload the only W1 rows that matter for this component.
    const float wc0 = W1[c  * HID + u0] + b1[u0];
    const float wc1 = W1[c  * HID + u1] + b1[u1];
    const float w80 = W1[8  * HID + u0], w81 = W1[8  * HID + u1];
    const float w90 = W1[9  * HID + u0], w91 = W1[9  * HID + u1];
    const float wa0 = W1[10 * HID + u0], wa1 = W1[10 * HID + u1];

    const unsigned char* sq = &seqb[c * NTIME];
    float acc0 = 0.0f, acc1 = 0.0f;
    int cnt = 0;

    for (int tb = 0; tb < NTIME; tb += 32) {
        const int t  = tb + lane;
        const int sv = sq[t];
        const int pv = (t == 0) ? 2 : sq[t - 1];
        const bool st = (sv != pv);                  // run start?
        float sn = 0.0f, dn = 0.0f;
        if (st) {
            int j = t + 1;                           // run length = dist to next change
            while (j < NTIME && sq[j] == sv) j++;
            sn = (float)t       * (1.0f / (float)(NTIME - 1));
            dn = (float)(j - t) * (1.0f / (float)NTIME);
        }
        unsigned mask = (unsigned)(__ballot(st) & 0xffffffffull);
        const float sf = (float)sv;
        while (mask) {                               // iterate only start lanes
            const int e = __ffs(mask) - 1;
            mask &= mask - 1;
            const float se  = __shfl(sf, e, 32);
            const float sne = __shfl(sn, e, 32);
            const float dne = __shfl(dn, e, 32);
            const float a0 = fmaf(se, w80, fmaf(sne, w90, fmaf(dne, wa0, wc0)));
            const float a1 = fmaf(se, w81, fmaf(sne, w91, fmaf(dne, wa1, wc1)));
            acc0 += fmaxf(a0, 0.0f);
            acc1 += fmaxf(a1, 0.0f);
            cnt++;
        }
    }

    accLDS[c][u0] = acc0;
    accLDS[c][u1] = acc1;
    if (lane == 0) cntLDS[c] = cnt;
    __syncthreads();

    if (tid < HID) {
        float s = 0.0f;
        for (int cc = 0; cc < NCOMP; cc++) s += accLDS[cc][tid];
        S[b * HID + tid] = s;
    }
    if (tid == 0) {
        int tot = 0;
        for (int cc = 0; cc < NCOMP; cc++) tot += cntLDS[cc];
        validv[b] = (float)((tot > 0) ? tot : 1);
    }
}

// ---------------------------------------------------------------------------
// FP32 WMMA helper: one 16x16 tile of a 64xK(64)x64 GEMM out of LDS.
// A fragment (16x4 f32): lanes 0-15 M=lane K={kk,kk+1}; lanes 16-31 K={kk+2,kk+3}.
// B fragment (4x16 f32): lanes 0-15 N=lane K={kk,kk+1}; lanes 16-31 K={kk+2,kk+3}.
// ---------------------------------------------------------------------------
__device__ __forceinline__ v8f gemm64_tile(const float* Asm, const float* Bsm,
                                           int m0, int n0, int ln, int hi, v8f c)
{
#pragma unroll
    for (int kk = 0; kk < 64; kk += 4) {
        v2f a, bb;
        a.x  = Asm[(m0 + ln) * AP + kk     + 2 * hi];
        a.y  = Asm[(m0 + ln) * AP + kk + 1 + 2 * hi];
        bb.x = Bsm[(kk     + 2 * hi) * AP + n0 + ln];
        bb.y = Bsm[(kk + 1 + 2 * hi) * AP + n0 + ln];
        c = __builtin_amdgcn_wmma_f32_16x16x4_f32(false, a, false, bb,
                                                  (short)0, c, false, false);
    }
    return c;
}

// ---------------------------------------------------------------------------
// Kernel 2: fused GEMM chain for the whole batch (single workgroup, 16 waves,
// one 16x16 output tile per wave):
//   T1  = (S/valid) @ W2 + b2                         (== "pooled")
//   T2  = relu(T1 @ P1[:64] + (valid/MAXE)*P1[64] + pb1)
//   out = T2 @ P2 + pb2
// ---------------------------------------------------------------------------
__global__ __launch_bounds__(512) void head_gemm_kernel(
    const float* __restrict__ S, const float* __restrict__ validv,
    const float* __restrict__ W2, const float* __restrict__ b2,
    const float* __restrict__ P1, const float* __restrict__ pb1,
    const float* __restrict__ P2, const float* __restrict__ pb2,
    float* __restrict__ out)
{
    __shared__ float Asm[64 * AP];
    __shared__ float Bsm[64 * AP];
    __shared__ float cntf[BATCH];

    const int tid  = threadIdx.x;
    const int lane = tid & 31;
    const int wid  = tid >> 5;              // 16 waves
    const int hi = (lane >> 4) & 1, ln = lane & 15;
    const int m0 = (wid >> 2) * 16;         // tile row (batch)
    const int n0 = (wid & 3) * 16;          // tile col (feature)

    // A = S / valid ; count feature
    for (int f = tid; f < 64 * 64; f += 512) {
        int m = f >> 6, n = f & 63;
        Asm[m * AP + n] = S[f] / validv[m];
    }
    if (tid < BATCH) cntf[tid] = validv[tid] * (1.0f / MAX_EVENTS_F);
    for (int f = tid; f < 64 * 64; f += 512)
        Bsm[(f >> 6) * AP + (f & 63)] = W2[f];
    __syncthreads();

    // ---- GEMM1: T1 = A @ W2 + b2 ----
    v8f c1;
    {
        const float bias = b2[n0 + ln];
#pragma unroll
        for (int i = 0; i < 8; i++) c1[i] = bias;
        c1 = gemm64_tile(Asm, Bsm, m0, n0, ln, hi, c1);
    }
    __syncthreads();                         // all reads of Asm done
#pragma unroll
    for (int i = 0; i < 8; i++)
        Asm[(m0 + i + 8 * hi) * AP + n0 + ln] = c1[i];
    __syncthreads();

    for (int f = tid; f < 64 * 64; f += 512)
        Bsm[(f >> 6) * AP + (f & 63)] = P1[f];   // first 64 rows of P1
    __syncthreads();

    // ---- GEMM2: T2 = relu(T1 @ P1[:64] + cnt*P1[64] + pb1) ----
    v8f c2;
    {
        const float p1last = P1[64 * 64 + n0 + ln];
        const float bias   = pb1[n0 + ln];
#pragma unroll
        for (int i = 0; i < 8; i++)
            c2[i] = fmaf(cntf[m0 + i + 8 * hi], p1last, bias);
        c2 = gemm64_tile(Asm, Bsm, m0, n0, ln, hi, c2);
    }
    __syncthreads();
#pragma unroll
    for (int i = 0; i < 8; i++)
        Asm[(m0 + i + 8 * hi) * AP + n0 + ln] = fmaxf(c2[i], 0.0f);
    __syncthreads();

    for (int f = tid; f < 64 * 64; f += 512)
        Bsm[(f >> 6) * AP + (f & 63)] = P2[f];
    __syncthreads();

    // ---- GEMM3: out = T2 @ P2 + pb2 ----
    v8f c3;
    {
        const float bias = pb2[n0 + ln];
#pragma unroll
        for (int i = 0; i < 8; i++) c3[i] = bias;
        c3 = gemm64_tile(Asm, Bsm, m0, n0, ln, hi, c3);
    }
#pragma unroll
    for (int i = 0; i < 8; i++)
        out[(m0 + i + 8 * hi) * OUTD + n0 + ln] = c3[i];
}

// ---------------------------------------------------------------------------
extern "C" void kernel_launch(void* const* d_in, const int* in_sizes, int n_in,
                              void* d_out, int out_size, void* d_ws, size_t ws_size,
                              hipStream_t stream)
{
    const float* x   = (const float*)d_in[0];
    const float* W1  = (const float*)d_in[1];
    const float* b1  = (const float*)d_in[2];
    const float* W2  = (const float*)d_in[3];
    const float* b2  = (const float*)d_in[4];
    const float* P1  = (const float*)d_in[5];
    const float* pb1 = (const float*)d_in[6];
    const float* P2  = (const float*)d_in[7];
    const float* pb2 = (const float*)d_in[8];

    float* S      = (float*)d_ws;            // [64][64] pooled relu sums
    float* validv = S + BATCH * HID;         // [64] event counts

    event_scan_kernel<<<BATCH, 256, 0, stream>>>(x, W1, b1, S, validv);
    head_gemm_kernel<<<1, 512, 0, stream>>>(S, validv, W2, b2, P1, pb1, P2, pb2,
                                            (float*)d_out);
}